// FastAttention_60816736911454
// MI455X (gfx1250) — compile-verified
//
#include <hip/hip_runtime.h>

// ---------------- problem constants ----------------
#define B_      2
#define H_      8
#define NH      16      // B*H flattened heads
#define N_      4096
#define D_      64
#define M_      256
#define COND_   512
#define CHUNK_  128
#define NSEG    29      // 1 cond segment (512 rows) + 28 causal chunks
#define NCHK    32      // 4 cond output chunks + 28 causal output chunks

#define NORMALIZER 0.35355339059327379f   // 64^-0.25
#define RATIO      0.0625f                // 256^-0.5
#define KEPS       1e-4f
#define EPSC       1e-6f

// padded LDS row strides: even (b64-aligned frags) and not 0 mod 64 (no bank conflicts)
#define XS   68     // feats: x rows
#define PS   68     // feats: proj rows
#define KTS  66     // segsum: k'^T rows (m-major)
#define VTS  66     // segsum: v^T rows (e-major)
#define OKS  260    // apply: k' rows (row-major, 64 rows)
#define OVS  130    // apply: v^T rows (e-major, 128 cols)
#define OSS  66     // apply: masked-score rows

typedef __attribute__((ext_vector_type(2))) float v2f;
typedef __attribute__((ext_vector_type(8))) float v8f;

// CDNA5 f32 WMMA: D(16x16) = A(16x4) * B(4x16) + C
__device__ __forceinline__ v8f wmma4(v2f a, v2f b, v8f c) {
  return __builtin_amdgcn_wmma_f32_16x16x4_f32(false, a, false, b, (short)0, c, false, false);
}
__device__ __forceinline__ v8f vzero8() {
  v8f z = {0.f,0.f,0.f,0.f,0.f,0.f,0.f,0.f};
  return z;
}
// order-preserving float<->uint for atomicMax over signed floats
__device__ __forceinline__ unsigned f2ord(float f) {
  unsigned u = __float_as_uint(f);
  return (u & 0x80000000u) ? ~u : (u | 0x80000000u);
}
__device__ __forceinline__ float ord2f(unsigned u) {
  return (u & 0x80000000u) ? __uint_as_float(u & 0x7FFFFFFFu) : __uint_as_float(~u);
}

__global__ void init_kernel(unsigned* g) { *g = 0u; }

// ============================================================================
// Feature map: dash = (norm*X) @ proj^T via WMMA; 16 rows x 256 feats per block.
// IS_Q: per-row max -> q' = ratio*(exp(dash - diag - m) + keps)
// !IS_Q: raw dash + per-row diag out, atomicMax global max (exp applied later)
// ============================================================================
template<bool IS_Q>
__global__ void __launch_bounds__(256, 1)
feats_kernel(const float* __restrict__ X, const float* __restrict__ proj,
             float* __restrict__ dst, float* __restrict__ diag_out,
             unsigned* __restrict__ gmax) {
  extern __shared__ float lds[];
  float* lds_x    = lds;                 // 16*68  = 1088
  float* lds_p    = lds_x + 16*XS;       // 256*68 = 17408
  float* lds_dash = lds_p + 256*PS;      // 16*256 = 4096
  float* lds_red  = lds_dash + 4096;     // 256
  float* lds_diag = lds_red + 256;       // 16
  float* lds_rmax = lds_diag + 16;       // 16

  const int tid = threadIdx.x;
  const int bh = blockIdx.x >> 8;
  const int n0 = (blockIdx.x & 255) << 4;

  { // stage 16 rows of X (pre-scaled); padded rows, float4 aligned (68 = 17*4)
    float4 vv = ((const float4*)(X + ((size_t)bh*N_ + n0)*D_))[tid];
    vv.x *= NORMALIZER; vv.y *= NORMALIZER; vv.z *= NORMALIZER; vv.w *= NORMALIZER;
    ((float4*)lds_x)[(tid >> 4)*17 + (tid & 15)] = vv;
  }
  for (int i = tid; i < 4096; i += 256)
    ((float4*)lds_p)[(i >> 4)*17 + (i & 15)] = ((const float4*)proj)[i];
  __syncthreads();

  const int lane = tid & 31, wid = tid >> 5, half = lane >> 4, li = lane & 15;
#pragma unroll
  for (int t = 0; t < 2; ++t) {
    const int mt = wid*2 + t;
    v8f acc = vzero8();
    const float* pa = lds_x + li*XS + 2*half;             // A[i][k] row-contig
    const float* pb = lds_p + (mt*16 + li)*PS + 2*half;   // B[k][m] = proj[m][k]
#pragma unroll 4
    for (int ks = 0; ks < 16; ++ks) {
      v2f a = *(const v2f*)(pa + ks*4);
      v2f b = *(const v2f*)(pb + ks*4);
      acc = wmma4(a, b, acc);
    }
#pragma unroll
    for (int r = 0; r < 8; ++r)
      lds_dash[(r + 8*half)*256 + mt*16 + li] = acc[r];
  }
  __syncthreads();

  if (tid < 16) {                             // diag = 0.5 * sum((norm*x)^2)
    float s = 0.f;
    for (int j = 0; j < 64; ++j) { float xv = lds_x[tid*XS + j]; s += xv*xv; }
    lds_diag[tid] = 0.5f * s;
  }
  const int row = tid >> 4, seg = tid & 15;
  float pm = -1e30f;
  for (int j = 0; j < 16; ++j) pm = fmaxf(pm, lds_dash[row*256 + seg*16 + j]);
  lds_red[tid] = pm;
  __syncthreads();

  if (IS_Q) {
    if (tid < 16) {
      float mx = -1e30f;
      for (int j = 0; j < 16; ++j) mx = fmaxf(mx, lds_red[tid*16 + j]);
      lds_rmax[tid] = mx;
    }
    __syncthreads();
    const float dg = lds_diag[row] + lds_rmax[row];
    float* o = dst + ((size_t)bh*N_ + n0 + row)*M_ + seg*16;
    for (int j = 0; j < 16; ++j)
      o[j] = RATIO * (__expf(lds_dash[row*256 + seg*16 + j] - dg) + KEPS);
  } else {
    if (tid == 0) {
      float mx = -1e30f;
      for (int j = 0; j < 256; ++j) mx = fmaxf(mx, lds_red[j]);
      atomicMax(gmax, f2ord(mx));
    }
    if (tid < 16) diag_out[(size_t)bh*N_ + n0 + tid] = lds_diag[tid];
    float* o = dst + ((size_t)bh*N_ + n0 + row)*M_ + seg*16;
    for (int j = 0; j < 16; ++j)
      o[j] = lds_dash[row*256 + seg*16 + j];
  }
}

// k' = ratio*(exp(dash - diag - gmax) + keps), in place
__global__ void expk_kernel(float* __restrict__ kp, const float* __restrict__ kdiag,
                            const unsigned* __restrict__ gmax) {
  const float m = ord2f(*gmax);
  const size_t i = (size_t)blockIdx.x*256 + threadIdx.x;
  float4 vv = ((float4*)kp)[i];
  const float dg = kdiag[i >> 6] + m;
  vv.x = RATIO*(__expf(vv.x - dg) + KEPS);
  vv.y = RATIO*(__expf(vv.y - dg) + KEPS);
  vv.z = RATIO*(__expf(vv.z - dg) + KEPS);
  vv.w = RATIO*(__expf(vv.w - dg) + KEPS);
  ((float4*)kp)[i] = vv;
}

// ============================================================================
// Segment sums: ctx_s[256x64] = k'^T v, ksum_s[256] = col-sums of k'.
// k'/v are staged TRANSPOSED (m-major / e-major) so WMMA A/B frags are single
// conflict-free ds_load_b64 with immediate offsets. 464 parallel blocks.
// ============================================================================
__global__ void __launch_bounds__(256, 1)
segsum_kernel(const float* __restrict__ kp, const float* __restrict__ v,
              float* __restrict__ ctxseg, float* __restrict__ ksumseg) {
  extern __shared__ float lds[];
  float* kT = lds;              // 256 * 66 = 16896  (kT[m][n])
  float* vT = kT + 256*KTS;     //  64 * 66 = 4224   (vT[e][n])
  const int head = blockIdx.x / NSEG;
  const int sg   = blockIdx.x % NSEG;
  const int tid = threadIdx.x, lane = tid & 31, wid = tid >> 5, half = lane >> 4, li = lane & 15;
  const int nhalves = (sg == 0) ? 8 : 2;                  // 64-row stages
  const int nstart  = (sg == 0) ? 0 : (COND_ + (sg-1)*CHUNK_);
  const float* kph = kp + (size_t)head*N_*M_;
  const float* vh  = v  + (size_t)head*N_*D_;

  v8f acc[2][4];
#pragma unroll
  for (int t = 0; t < 2; ++t)
#pragma unroll
    for (int e = 0; e < 4; ++e) acc[t][e] = vzero8();
  float ks_acc = 0.f;

  for (int hh = 0; hh < nhalves; ++hh) {
    const int nb1 = nstart + hh*64;
    __syncthreads();
    for (int i = tid; i < 4096; i += 256) {          // k' 64x256 -> kT (transpose)
      const int n = i >> 6, c = (i & 63) << 2;
      float4 vv = ((const float4*)(kph + (size_t)nb1*M_))[i];
      kT[(c+0)*KTS + n] = vv.x; kT[(c+1)*KTS + n] = vv.y;
      kT[(c+2)*KTS + n] = vv.z; kT[(c+3)*KTS + n] = vv.w;
    }
    for (int i = tid; i < 1024; i += 256) {          // v 64x64 -> vT (transpose)
      const int n = i >> 4, c = (i & 15) << 2;
      float4 vv = ((const float4*)(vh + (size_t)nb1*D_))[i];
      vT[(c+0)*VTS + n] = vv.x; vT[(c+1)*VTS + n] = vv.y;
      vT[(c+2)*VTS + n] = vv.z; vT[(c+3)*VTS + n] = vv.w;
    }
    __syncthreads();
#pragma unroll
    for (int t = 0; t < 2; ++t) {
      const int mt = wid*2 + t;                       // wave owns m in [32w,32w+32)
      const float* ka = kT + (mt*16 + li)*KTS + 2*half;
#pragma unroll
      for (int et = 0; et < 4; ++et) {
        const float* vb = vT + (et*16 + li)*VTS + 2*half;
        v8f a8 = acc[t][et];
#pragma unroll 4
        for (int ks = 0; ks < 16; ++ks) {
          v2f a = *(const v2f*)(ka + ks*4);           // A[m][n] = k'[n][m]
          v2f b = *(const v2f*)(vb + ks*4);           // B[n][e] = v[n][e]
          a8 = wmma4(a, b, a8);
        }
        acc[t][et] = a8;
      }
    }
    { float s = 0.f; for (int n2 = 0; n2 < 64; ++n2) s += kT[tid*KTS + n2]; ks_acc += s; }
  }

  float* cdst = ctxseg + ((size_t)head*NSEG + sg)*16384;   // ctx[m][e] row-major
#pragma unroll
  for (int t = 0; t < 2; ++t) {
    const int mt = wid*2 + t;
#pragma unroll
    for (int et = 0; et < 4; ++et)
#pragma unroll
      for (int r = 0; r < 8; ++r)
        cdst[(mt*16 + r + 8*half)*64 + et*16 + li] = acc[t][et][r];
  }
  ksumseg[((size_t)head*NSEG + sg)*256 + tid] = ks_acc;
}

// Inclusive prefix over segments per head; fold the single cond EPS into ksum[0].
__global__ void scan_kernel(float* __restrict__ ctxseg, float* __restrict__ ksumseg) {
  const int head = blockIdx.x, tid = threadIdx.x;
  float* cbase = ctxseg  + (size_t)head*NSEG*16384;
  float* kbase = ksumseg + (size_t)head*NSEG*256;
  float run = kbase[tid] + EPSC;
  kbase[tid] = run;
  for (int c = 1; c < NSEG; ++c) { run += kbase[c*256 + tid]; kbase[c*256 + tid] = run; }
  for (int i = tid; i < 16384; i += 256) {   // same-thread element chain
    float r = cbase[i];
    for (int c = 1; c < NSEG; ++c) { r += cbase[c*16384 + i]; cbase[c*16384 + i] = r; }
  }
}

// ============================================================================
// Apply: one 128-row output chunk per block (512 blocks, 8 waves = 8 row-tiles).
//  cond (cc<4):  out = (q' @ ctx[0]) / (q' . ksum[0])
//  causal:       out = (tril(q'k'^T)v + q' @ ctx[c]) /
//                      (q'.ksum[c] + eps*rowsum(q') + rowsum(tril(scores)))
// ============================================================================
__global__ void __launch_bounds__(256, 1)
out_kernel(const float* __restrict__ qp, const float* __restrict__ kp,
           const float* __restrict__ v, const float* __restrict__ ctxseg,
           const float* __restrict__ ksumseg, float* __restrict__ out) {
  extern __shared__ float lds[];
  float* ksum = lds;              // 256
  float* dden = ksum + 256;       // 128
  float* rsum = dden + 128;       // 128
  float* vT   = rsum + 128;       // 64*130 = 8320   (vT[e][n], n in [0,128))
  float* sbuf = vT + 64*OVS;      // 8*16*66 = 8448  (per-wave masked scores)
  float* kbuf = sbuf + 8*16*OSS;  // 64*260 = 16640  (k' half-chunk, row-major)

  const int head = blockIdx.x >> 5;
  const int cc   = blockIdx.x & 31;
  const bool causal = (cc >= 4);
  const int st    = causal ? (cc - 4) : 0;
  const int nbase = causal ? (COND_ + (cc - 4)*CHUNK_) : cc*CHUNK_;
  const int tid = threadIdx.x, lane = tid & 31, wid = tid >> 5, half = lane >> 4, li = lane & 15;

  const float* qph  = qp + (size_t)head*N_*M_;
  const float* kph  = kp + (size_t)head*N_*M_;
  const float* vh   = v  + (size_t)head*N_*D_;
  const float* ctxg = ctxseg + ((size_t)head*NSEG + st)*16384;
  float* oh = out + (size_t)head*N_*D_;

  ksum[tid] = ksumseg[((size_t)head*NSEG + st)*256 + tid];
  if (causal)
    for (int i = tid; i < 2048; i += 256) {        // v 128x64 -> vT (transpose)
      const int n = i >> 4, c = (i & 15) << 2;
      float4 vv = ((const float4*)(vh + (size_t)nbase*D_))[i];
      vT[(c+0)*OVS + n] = vv.x; vT[(c+1)*OVS + n] = vv.y;
      vT[(c+2)*OVS + n] = vv.z; vT[(c+3)*OVS + n] = vv.w;
    }
  __syncthreads();

  { // denominator base: 2 lanes per row, each 128 features (float4)
    const int rloc = lane >> 1, hseg = lane & 1;
    const float4* qr4 = (const float4*)(qph + (size_t)(nbase + wid*16 + rloc)*M_ + hseg*128);
    const float* ks4 = ksum + hseg*128;
    float dot = 0.f, rsq = 0.f;
    for (int f = 0; f < 32; ++f) {
      const float4 qv = qr4[f];
      dot += qv.x*ks4[f*4+0] + qv.y*ks4[f*4+1] + qv.z*ks4[f*4+2] + qv.w*ks4[f*4+3];
      rsq += qv.x + qv.y + qv.z + qv.w;
    }
    dot += __shfl_xor(dot, 1);
    rsq += __shfl_xor(rsq, 1);
    if (hseg == 0) dden[wid*16 + rloc] = causal ? (dot + EPSC*rsq) : dot;
  }

  v8f acc[4];
#pragma unroll
  for (int e = 0; e < 4; ++e) acc[e] = vzero8();

  // inter: q'(16xM) @ ctx(Mx64); ctx row-major global (lane-coalesced b32 pairs)
  {
    const float* qa = qph + (size_t)(nbase + wid*16 + li)*M_ + 2*half;
#pragma unroll 2
    for (int ks = 0; ks < 64; ++ks) {
      const int fb = ks*4 + 2*half;
      v2f a = *(const v2f*)(qa + ks*4);
#pragma unroll
      for (int et = 0; et < 4; ++et) {
        v2f b; b.x = ctxg[fb*64 + et*16 + li]; b.y = ctxg[(fb+1)*64 + et*16 + li];
        acc[et] = wmma4(a, b, acc[et]);
      }
    }
  }

  if (causal) {
    float rs_acc[8] = {0.f,0.f,0.f,0.f,0.f,0.f,0.f,0.f};
    for (int jh = 0; jh < 2; ++jh) {
      __syncthreads();
      for (int i = tid; i < 4096; i += 256)        // stage 64 k' rows (padded)
        ((float4*)kbuf)[(i >> 6)*65 + (i & 63)] =
            ((const float4*)(kph + (size_t)(nbase + jh*64)*M_))[i];
      __syncthreads();
      v8f sc[4];
#pragma unroll
      for (int jt = 0; jt < 4; ++jt) sc[jt] = vzero8();
      {
        const float* qa = qph + (size_t)(nbase + wid*16 + li)*M_ + 2*half;
#pragma unroll 2
        for (int ks = 0; ks < 64; ++ks) {          // scores = q' k'^T (256 feats)
          v2f a = *(const v2f*)(qa + ks*4);
#pragma unroll
          for (int jt = 0; jt < 4; ++jt) {
            v2f b = *(const v2f*)(kbuf + (jt*16 + li)*OKS + 2*half + ks*4);
            sc[jt] = wmma4(a, b, sc[jt]);
          }
        }
      }
#pragma unroll
      for (int r = 0; r < 8; ++r) {                // tril mask + rowsum + stash
        const int irow = wid*16 + r + 8*half;
#pragma unroll
        for (int jt = 0; jt < 4; ++jt) {
          const int j = jh*64 + jt*16 + li;
          float xv = sc[jt][r];
          xv = (j <= irow) ? xv : 0.f;
          rs_acc[r] += xv;
          sbuf[wid*16*OSS + (r + 8*half)*OSS + jt*16 + li] = xv;
        }
      }
      {
        const float* sa = sbuf + wid*16*OSS + li*OSS + 2*half;
#pragma unroll 4
        for (int ks = 0; ks < 16; ++ks) {          // intra: tril(scores) @ v
          const int jb = ks*4 + 2*half;
          v2f a = *(const v2f*)(sa + ks*4);
#pragma unroll
          for (int et = 0; et < 4; ++et) {
            v2f b = *(const v2f*)(vT + (et*16 + li)*OVS + jh*64 + jb);
            acc[et] = wmma4(a, b, acc[et]);
          }
        }
      }
    }
#pragma unroll
    for (int r = 0; r < 8; ++r) {                  // reduce rowsums over 16 lanes
      float rs = rs_acc[r];
      rs += __shfl_xor(rs, 1); rs += __shfl_xor(rs, 2);
      rs += __shfl_xor(rs, 4); rs += __shfl_xor(rs, 8);
      if (li == 0) rsum[wid*16 + r + 8*half] = rs;
    }
  }

#pragma unroll
  for (int r = 0; r < 8; ++r) {
    const int row = wid*16 + r + 8*half;
    const float den = causal ? (dden[row] + rsum[row]) : dden[row];
    const float dv = 1.0f / den;
    const int n = nbase + row;
#pragma unroll
    for (int et = 0; et < 4; ++et)
      oh[(size_t)n*64 + et*16 + li] = acc[et][r]*dv;
  }
}

// ============================================================================
extern "C" void kernel_launch(void* const* d_in, const int* in_sizes, int n_in,
                              void* d_out, int out_size, void* d_ws, size_t ws_size,
                              hipStream_t stream) {
  const float* q    = (const float*)d_in[0];
  const float* k    = (const float*)d_in[1];
  const float* v    = (const float*)d_in[2];
  const float* proj = (const float*)d_in[3];
  float* out = (float*)d_out;

  float* qp      = (float*)d_ws;                          // 16 * 4096 * 256
  float* kp      = qp      + (size_t)NH*N_*M_;            // 16 * 4096 * 256
  float* kdiag   = kp      + (size_t)NH*N_*M_;            // 16 * 4096
  float* ctxseg  = kdiag   + (size_t)NH*N_;               // 16 * 29 * 256*64
  float* ksumseg = ctxseg  + (size_t)NH*NSEG*16384;       // 16 * 29 * 256
  unsigned* gmax = (unsigned*)(ksumseg + (size_t)NH*NSEG*256);

  const size_t featsh = (size_t)(16*XS + 256*PS + 4096 + 256 + 16 + 16) * sizeof(float); // ~89 KB
  const size_t segsh  = (size_t)(256*KTS + 64*VTS) * sizeof(float);                      // ~83 KB
  const size_t outsh  = (size_t)(256 + 128 + 128 + 64*OVS + 8*16*OSS + 64*OKS) * sizeof(float); // ~133 KB

  init_kernel<<<1, 1, 0, stream>>>(gmax);
  feats_kernel<true ><<<dim3(NH*(N_/16)), dim3(256), featsh, stream>>>(q, proj, qp, nullptr, nullptr);
  feats_kernel<false><<<dim3(NH*(N_/16)), dim3(256), featsh, stream>>>(k, proj, kp, kdiag, gmax);
  expk_kernel<<<dim3((NH*N_*M_)/1024), dim3(256), 0, stream>>>(kp, kdiag, gmax);
  segsum_kernel<<<dim3(NH*NSEG), dim3(256), segsh, stream>>>(kp, v, ctxseg, ksumseg);
  scan_kernel<<<dim3(NH), dim3(256), 0, stream>>>(ctxseg, ksumseg);
  out_kernel<<<dim3(NH*NCHK), dim3(256), outsh, stream>>>(qp, kp, v, ctxseg, ksumseg, out);
}